// BasicTT_13649406066761
// MI455X (gfx1250) — compile-verified
//
#include <hip/hip_runtime.h>
#include <hip/hip_bf16.h>

// CDNA5 / gfx1250, wave32. All matrix work via V_WMMA_F32_16X16X4_F32.
typedef float v2f __attribute__((ext_vector_type(2)));
typedef float v8f __attribute__((ext_vector_type(8)));

constexpr int BATCH = 512;

// ---------------------------------------------------------------------------
// One-time core packing into the ISA f32 16x4 A-operand lane layout:
//   packed[(k4*32 + lane)*2 + j] = core[K=(k4*4 + 2*half + j)][M=lane%16]
// with zeros baked in for lanes >= M and K-chunks past K. Runs once per core.
// ---------------------------------------------------------------------------
template<int R, int N, int MM, int SS>
__global__ void pack_core_kernel(const float* __restrict__ core,
                                 float* __restrict__ packed)
{
    constexpr int K = R * N, M = MM * SS, K4 = (K + 3) / 4;
    const int lane = threadIdx.x & 31;
    const int half = lane >> 4, l16 = lane & 15;
#pragma unroll
    for (int k4 = 0; k4 < K4; ++k4) {
#pragma unroll
        for (int j = 0; j < 2; ++j) {
            const int kk = k4 * 4 + 2 * half + j;
            float av = 0.f;
            if (kk < K && l16 < M) {
                const int ri = kk / N,  ni = kk - ri * N;
                const int mi = l16 / SS, si = l16 - mi * SS;
                av = core[((ri * MM + mi) * N + ni) * SS + si];
            }
            packed[(k4 * 32 + lane) * 2 + j] = av;
        }
    }
}

// Pack w_lin [10,64] the same way (16 K-chunks, zero rows 10..15).
__global__ void pack_w_kernel(const float* __restrict__ w,
                              float* __restrict__ packed)
{
    const int lane = threadIdx.x & 31;
    const int half = lane >> 4, l16 = lane & 15;
#pragma unroll
    for (int k4 = 0; k4 < 16; ++k4) {
#pragma unroll
        for (int j = 0; j < 2; ++j) {
            const int kk = k4 * 4 + 2 * half + j;
            packed[(k4 * 32 + lane) * 2 + j] = (l16 < 10) ? w[l16 * 64 + kk] : 0.f;
        }
    }
}

// ---------------------------------------------------------------------------
// TT einsum step, fully constant-specialized (R,N,MM,SS,A,Q all compile-time):
//   zin  : [B, A, R, N, Q]   zout : [B, A, MM, SS, Q]
// One wave per (b, a, 16-wide q tile). A fragments: unconditional b64 loads
// from the packed core. B fragments: unconditional b32 loads with clamped
// indices (clamped columns/rows never contribute: not stored / zero A cols).
// EXEC is all-ones at every WMMA.
// ---------------------------------------------------------------------------
template<int R, int N, int MM, int SS, int A, int Q, bool RELU>
__global__ __launch_bounds__(256) void tt_step_kernel(
        const float* __restrict__ zin, const float* __restrict__ pcore,
        const float* __restrict__ bias, float* __restrict__ zout)
{
    constexpr int  K      = R * N;           // 2..24
    constexpr int  M      = MM * SS;         // <= 12
    constexpr int  K4     = (K + 3) / 4;
    constexpr bool QG     = (Q & 15) != 0;   // only tail steps (Q in {8,4,1})
    constexpr int  qTiles = (Q + 15) / 16;
    constexpr int  tiles  = BATCH * A * qTiles;

    const int lane = threadIdx.x & 31;
    const int wave = blockIdx.x * (blockDim.x >> 5) + (threadIdx.x >> 5);
    if (wave >= tiles) return;               // wave-uniform: EXEC stays full

    int t = wave;
    const int qt = t % qTiles;  t /= qTiles; // constant divisors -> mul/shift
    const int a  = t % A;       t /= A;
    const int b  = t;

    const int half = lane >> 4;              // K-pair selector (ISA f32 layout)
    const int l16  = lane & 15;
    const int q    = (qt << 4) + l16;
    const int qc   = QG ? (q < Q ? q : Q - 1) : q;  // clamp: never stored

    const int inOff  = (b * A + a) * (K * Q);       // < 2^31 for all steps
    const int outOff = (b * A + a) * (M * Q);

    // ---- A fragments: unconditional, coalesced, L2-hot ----
    v2f afrag[K4];
#pragma unroll
    for (int k4 = 0; k4 < K4; ++k4)
        afrag[k4] = *(const v2f*)(pcore + (k4 * 32 + lane) * 2);

    // ---- accumulate ----
    v8f acc = {0.f, 0.f, 0.f, 0.f, 0.f, 0.f, 0.f, 0.f};
#pragma unroll
    for (int k4 = 0; k4 < K4; ++k4) {
        v2f bfrag;
#pragma unroll
        for (int j = 0; j < 2; ++j) {
            int kk = k4 * 4 + 2 * half + j;
            if (K & 3) kk = (kk < K) ? kk : (K - 1);  // only K=2 step; A col zero
            bfrag[j] = zin[inOff + kk * Q + qc];
        }
        acc = __builtin_amdgcn_wmma_f32_16x16x4_f32(
            /*neg_a=*/false, afrag[k4], /*neg_b=*/false, bfrag,
            /*c_mod=*/(short)0, acc, /*reuse_a=*/false, /*reuse_b=*/false);
    }

    // ---- store ----
    if (!QG || q < Q) {
#pragma unroll
        for (int v = 0; v < 8; ++v) {
            const int row = v + 8 * half;    // C/D layout: vgpr v -> M row
            if (row < M) {                   // folds: constant or half-test
                float val = acc[v];
                if (RELU) {                  // last step: SS==1, Q==1
                    val += bias[a * MM + row];
                    val = fmaxf(val, 0.f);
                }
                zout[outOff + row * Q + q] = val;
            }
        }
    }
}

// ---------------------------------------------------------------------------
// Final head: out[b, j] = sum_k h[b,k] w[j,k] + bl[j]. 512 % 16 == 0 -> no
// batch guards. M = 10 rows (packed w zero-padded), K = 64 -> 16 WMMAs/wave.
// ---------------------------------------------------------------------------
__global__ __launch_bounds__(256) void linear_kernel(
        const float* __restrict__ h, const float* __restrict__ pw,
        const float* __restrict__ bl, float* __restrict__ out)
{
    const int lane = threadIdx.x & 31;
    const int wave = blockIdx.x * (blockDim.x >> 5) + (threadIdx.x >> 5);
    if (wave >= BATCH / 16) return;

    const int half = lane >> 4, l16 = lane & 15;
    const int b = (wave << 4) + l16;

    v8f acc = {0.f, 0.f, 0.f, 0.f, 0.f, 0.f, 0.f, 0.f};
#pragma unroll
    for (int k4 = 0; k4 < 16; ++k4) {
        v2f afrag = *(const v2f*)(pw + (k4 * 32 + lane) * 2);
        v2f bfrag;
#pragma unroll
        for (int j = 0; j < 2; ++j) {
            const int kk = k4 * 4 + 2 * half + j;
            bfrag[j] = h[b * 64 + kk];
        }
        acc = __builtin_amdgcn_wmma_f32_16x16x4_f32(
            false, afrag, false, bfrag, (short)0, acc, false, false);
    }

#pragma unroll
    for (int v = 0; v < 8; ++v) {
        const int row = v + 8 * half;
        if (row < 10) out[b * 10 + row] = acc[v] + bl[row];
    }
}

// ---------------------------------------------------------------------------
// Host side
// ---------------------------------------------------------------------------
template<int R, int N, int MM, int SS>
static void pack(const void* core, float* packed, hipStream_t s)
{
    pack_core_kernel<R, N, MM, SS><<<dim3(1), dim3(32), 0, s>>>(
        (const float*)core, packed);
}

template<int R, int N, int MM, int SS, int A, int Q, bool RELU>
static void step(const float* zin, const float* pcore, const float* bias,
                 float* zout, hipStream_t s)
{
    constexpr int qTiles = (Q + 15) / 16;
    constexpr int tiles  = BATCH * A * qTiles;
    constexpr int blocks = (tiles + 7) / 8;          // 8 wave32 per block
    tt_step_kernel<R, N, MM, SS, A, Q, RELU>
        <<<dim3(blocks), dim3(256), 0, s>>>(zin, pcore, bias, zout);
}

extern "C" void kernel_launch(void* const* d_in, const int* in_sizes, int n_in,
                              void* d_out, int out_size, void* d_ws, size_t ws_size,
                              hipStream_t stream)
{
    // Input order (pytree-flattened setup_inputs dict):
    //  0: x | 1..6: cores1 | 7: bias1 | 8..13: cores2 | 14: bias2
    //  15..20: cores3 | 21: bias3 | 22: w_lin | 23: b_lin
    const float* x     = (const float*)d_in[0];
    const float* bias1 = (const float*)d_in[7];
    const float* bias2 = (const float*)d_in[14];
    const float* bias3 = (const float*)d_in[21];
    const float* w_lin = (const float*)d_in[22];
    const float* b_lin = (const float*)d_in[23];
    float* out = (float*)d_out;

    // Workspace (floats). Peak ~311 MB; every intermediate is L2-resident on
    // MI455X (largest single buffer 192 MB <= 192 MB L2).
    float* ws = (float*)d_ws;
    const long long BUFA_ELEMS = 50331648LL;  // TT1 step-0 out: 512*4*3*8192
    const long long BUFB_ELEMS = 25165824LL;  // TT1 step-1 out: 512*16*3*1024
    const long long H1_ELEMS   = (long long)BATCH * 4096;
    const long long H2_ELEMS   = (long long)BATCH * 256;
    const long long H3_ELEMS   = (long long)BATCH * 64;

    float* bufA = ws;
    float* bufB = bufA + BUFA_ELEMS;
    float* h1   = bufB + BUFB_ELEMS;
    float* h2   = h1 + H1_ELEMS;
    float* h3   = h2 + H2_ELEMS;
    float* pk   = h3 + H3_ELEMS;              // 19 packed slots x 1024 floats
    (void)ws_size; (void)in_sizes; (void)n_in; (void)out_size;

    // ---- pack all cores + w_lin into WMMA A-fragment layout (one-time) ----
    pack<1, 4, 4, 3>(d_in[1],  pk + 0  * 1024, stream);
    pack<3, 8, 4, 3>(d_in[2],  pk + 1  * 1024, stream);
    pack<3, 8, 4, 3>(d_in[3],  pk + 2  * 1024, stream);
    pack<3, 4, 4, 3>(d_in[4],  pk + 3  * 1024, stream);
    pack<3, 8, 4, 3>(d_in[5],  pk + 4  * 1024, stream);
    pack<3, 8, 4, 1>(d_in[6],  pk + 5  * 1024, stream);
    pack<1, 4, 2, 2>(d_in[8],  pk + 6  * 1024, stream);
    pack<2, 4, 2, 2>(d_in[9],  pk + 7  * 1024, stream);
    pack<2, 4, 4, 2>(d_in[10], pk + 8  * 1024, stream);
    pack<2, 4, 2, 2>(d_in[11], pk + 9  * 1024, stream);
    pack<2, 4, 2, 2>(d_in[12], pk + 10 * 1024, stream);
    pack<2, 4, 4, 1>(d_in[13], pk + 11 * 1024, stream);
    pack<1, 2, 2, 2>(d_in[15], pk + 12 * 1024, stream);
    pack<2, 2, 2, 2>(d_in[16], pk + 13 * 1024, stream);
    pack<2, 4, 2, 2>(d_in[17], pk + 14 * 1024, stream);
    pack<2, 2, 2, 2>(d_in[18], pk + 15 * 1024, stream);
    pack<2, 2, 2, 2>(d_in[19], pk + 16 * 1024, stream);
    pack<2, 4, 2, 1>(d_in[20], pk + 17 * 1024, stream);
    pack_w_kernel<<<dim3(1), dim3(32), 0, stream>>>(w_lin, pk + 18 * 1024);

    // ---- TT layer 1: m=[4x6], n=[4,8,8,4,8,8], r=[1,3,3,3,3,3,1] ----
    step<1, 4, 4, 3, 1,    8192, false>(x,    pk + 0 * 1024, nullptr, bufA, stream);
    step<3, 8, 4, 3, 4,    1024, false>(bufA, pk + 1 * 1024, nullptr, bufB, stream);
    step<3, 8, 4, 3, 16,   128,  false>(bufB, pk + 2 * 1024, nullptr, bufA, stream);
    step<3, 4, 4, 3, 64,   32,   false>(bufA, pk + 3 * 1024, nullptr, bufB, stream);
    step<3, 8, 4, 3, 256,  4,    false>(bufB, pk + 4 * 1024, nullptr, bufA, stream);
    step<3, 8, 4, 1, 1024, 1,    true >(bufA, pk + 5 * 1024, bias1,   h1,   stream);

    // ---- TT layer 2: m=[2,2,4,2,2,4], n=[4x6], r=[1,2,2,2,2,2,1] ----
    step<1, 4, 2, 2, 1,  1024, false>(h1,   pk + 6  * 1024, nullptr, bufA, stream);
    step<2, 4, 2, 2, 2,  256,  false>(bufA, pk + 7  * 1024, nullptr, bufB, stream);
    step<2, 4, 4, 2, 4,  64,   false>(bufB, pk + 8  * 1024, nullptr, bufA, stream);
    step<2, 4, 2, 2, 16, 16,   false>(bufA, pk + 9  * 1024, nullptr, bufB, stream);
    step<2, 4, 2, 2, 32, 4,    false>(bufB, pk + 10 * 1024, nullptr, bufA, stream);
    step<2, 4, 4, 1, 64, 1,    true >(bufA, pk + 11 * 1024, bias2,   h2,   stream);

    // ---- TT layer 3: m=[2x6], n=[2,2,4,2,2,4], r=[1,2,2,2,2,2,1] ----
    step<1, 2, 2, 2, 1,  128, false>(h2,   pk + 12 * 1024, nullptr, bufA, stream);
    step<2, 2, 2, 2, 2,  64,  false>(bufA, pk + 13 * 1024, nullptr, bufB, stream);
    step<2, 4, 2, 2, 4,  16,  false>(bufB, pk + 14 * 1024, nullptr, bufA, stream);
    step<2, 2, 2, 2, 8,  8,   false>(bufA, pk + 15 * 1024, nullptr, bufB, stream);
    step<2, 2, 2, 2, 16, 4,   false>(bufB, pk + 16 * 1024, nullptr, bufA, stream);
    step<2, 4, 2, 1, 32, 1,   true >(bufA, pk + 17 * 1024, bias3,   h3,   stream);

    // ---- Final linear head: 32 waves -> 4 blocks of 256 ----
    linear_kernel<<<dim3(4), dim3(256), 0, stream>>>(h3, pk + 18 * 1024, b_lin, out);
}